// Embedding_22505628631768
// MI455X (gfx1250) — compile-verified
//
#include <hip/hip_runtime.h>
#include <hip/hip_bf16.h>

// Embedding row-gather: out[row, :] = emb[x[row], :]
//   x:   [16384] int32 (4 x 4096 flattened)
//   emb: [50257, 1024] f32 (row = 4096 bytes)
//   out: [16384, 1024] f32
//
// Pure HBM-bandwidth problem (~134 MB => ~5.8 us floor at 23.3 TB/s).
// Uses the CDNA5 async LDS DMA path: global_load_async_to_lds_b128 /
// global_store_async_from_lds_b128 (ASYNCcnt), so row data never touches
// VGPRs. One wave stages one 4 KB row through a private LDS slot.

namespace {
constexpr int ROW_BYTES = 4096;   // 1024 f32 per embedding row
constexpr int WAVES     = 8;      // waves per block (wave32)
constexpr int THREADS   = WAVES * 32;
}

__global__ __launch_bounds__(THREADS)
void emb_gather_async_kernel(const int* __restrict__ x,
                             const float* __restrict__ emb,
                             float* __restrict__ out,
                             int n_rows)
{
    // 4 KB staging slot per wave; 32 KB/block of the 320 KB WGP pool.
    __shared__ __align__(128) char smem[WAVES * ROW_BYTES];

    const int wave = threadIdx.x >> 5;
    const int lane = threadIdx.x & 31;

    // Low 32 bits of a generic LDS pointer are the workgroup-relative LDS
    // byte offset — exactly what the async ops' LDS-address VGPR expects.
    const unsigned ldsaddr =
        (unsigned)(uintptr_t)(smem + wave * ROW_BYTES + lane * 16);

    const int stride = (int)gridDim.x * WAVES;

    for (int row = (int)blockIdx.x * WAVES + wave; row < n_rows; row += stride) {
        const int idx = x[row];   // wave-uniform gather index

        const char* src = (const char*)emb + (size_t)idx * ROW_BYTES + (size_t)(lane * 16);
        char*       dst = (char*)out       + (size_t)row * ROW_BYTES + (size_t)(lane * 16);

        // 8 x b128 per wave = 4 KB row. INST_OFFSET applies to BOTH the LDS
        // and global addresses (CDNA5 ISA), so one
        // base register pair covers all chunks.
#define EMB_ASYNC_LD(OFF)                                                      \
        asm volatile("global_load_async_to_lds_b128 %0, %1, off offset:" #OFF  \
                     :: "v"(ldsaddr), "v"(src) : "memory")
        EMB_ASYNC_LD(0);    EMB_ASYNC_LD(512);
        EMB_ASYNC_LD(1024); EMB_ASYNC_LD(1536);
        EMB_ASYNC_LD(2048); EMB_ASYNC_LD(2560);
        EMB_ASYNC_LD(3072); EMB_ASYNC_LD(3584);
#undef EMB_ASYNC_LD

        // Async load->store through the same LDS bytes is NOT ordered; drain
        // the loads before the stores read the staging slot.
        asm volatile("s_wait_asynccnt 0x0" ::: "memory");

#define EMB_ASYNC_ST(OFF)                                                       \
        asm volatile("global_store_async_from_lds_b128 %0, %1, off offset:" #OFF\
                     :: "v"(dst), "v"(ldsaddr) : "memory")
        EMB_ASYNC_ST(0);    EMB_ASYNC_ST(512);
        EMB_ASYNC_ST(1024); EMB_ASYNC_ST(1536);
        EMB_ASYNC_ST(2048); EMB_ASYNC_ST(2560);
        EMB_ASYNC_ST(3072); EMB_ASYNC_ST(3584);
#undef EMB_ASYNC_ST

        // Stores must drain before the next iteration's loads overwrite this
        // wave's staging slot. Per-wave serialization is fine: bandwidth is
        // hidden by ~16K concurrent waves, not intra-wave pipelining.
        asm volatile("s_wait_asynccnt 0x0" ::: "memory");
    }
}

extern "C" void kernel_launch(void* const* d_in, const int* in_sizes, int n_in,
                              void* d_out, int out_size, void* d_ws, size_t ws_size,
                              hipStream_t stream)
{
    (void)n_in; (void)out_size; (void)d_ws; (void)ws_size;

    const int*   x   = (const int*)d_in[0];     // [16384] int32
    const float* emb = (const float*)d_in[1];   // [50257, 1024] f32
    float*       out = (float*)d_out;           // [16384, 1024] f32

    const int n_rows = in_sizes[0];             // 16384
    const int blocks = (n_rows + WAVES - 1) / WAVES;  // 2048

    emb_gather_async_kernel<<<blocks, THREADS, 0, stream>>>(x, emb, out, n_rows);
}